// _StackedGCN_55568286876147
// MI455X (gfx1250) — compile-verified
//
#include <hip/hip_runtime.h>

// Stacked GCN (3x GCNConv + classifier) for MI455X / gfx1250.
//
// Cost model: GEMMs are ~2.9 GFLOP (trivial); the per-edge gather/scatter-add
// (3 x 1.6M edges x 64 feats x 8B) is ~2.5 GB of traffic and dominates. h/agg
// buffers (25.6 MB each) are L2-resident (192 MB L2), so we use hardware
// global_atomic_add_f32 at L2 for the scatter and keep the GEMM in full fp32
// via V_WMMA_F32_16X16X4_F32 (no precision loss, still a CDNA5 matrix op).
//
// Round-1 fix: GEMM is now templated on NCOLS/RELU so the accumulator tiles
// are statically indexed -> no v_movrels/v_movreld M0-indexed register
// shuffling around the WMMAs; fully unrolled K loop.

#define NN   100000
#define NE   1600000
#define KDIM 64
#define HDIM 64
#define ODIM 32

typedef __attribute__((ext_vector_type(2))) float v2f;
typedef __attribute__((ext_vector_type(8))) float v8f;

// ---------------- degree / normalization ----------------

__global__ void gcn_init_deg(float* __restrict__ deg, int n) {
    int i = blockIdx.x * blockDim.x + threadIdx.x;
    if (i < n) deg[i] = 1.0f;   // self-loop
}

__global__ void gcn_deg_accum(const int* __restrict__ dst, float* __restrict__ deg, int e) {
    int i = blockIdx.x * blockDim.x + threadIdx.x;
    if (i < e) unsafeAtomicAdd(&deg[dst[i]], 1.0f);
}

__global__ void gcn_deg_final(float* __restrict__ deg_dis, float* __restrict__ dinv, int n) {
    int i = blockIdx.x * blockDim.x + threadIdx.x;
    if (i < n) {
        float d = deg_dis[i];
        deg_dis[i] = rsqrtf(d);   // dis = deg^{-1/2}, in place
        dinv[i]    = 1.0f / d;    // self-loop coefficient
    }
}

// ---------------- WMMA GEMM: Y = (relu?)X @ W (+bias) ----------------
// One wave computes a 16-row strip of all NCOLS outputs. W staged in LDS.
// A (16x4 f32): lane r=lane&15 is row M=r; vgpr j holds K = kk + j + 2*(lane>>4).
// B (4x16 f32): lane r is col N=r;        vgpr j holds K = kk + j + 2*(lane>>4).
// C/D (16x16 f32): vgpr j, lane r -> row M = j + 8*(lane>>4), col N = r.

template <int NCOLS, bool RELU>
__global__ __launch_bounds__(256)
void gcn_gemm_wmma(const float* __restrict__ X, const float* __restrict__ W,
                   const float* __restrict__ bias, float* __restrict__ Y,
                   int nrows) {
    constexpr int NCT = NCOLS / 16;       // 16-col tiles (4 for H=64, 2 for OUT=32)
    __shared__ float wlds[KDIM * NCOLS];  // <= 16 KB

    int tid = threadIdx.x;
    #pragma unroll
    for (int i = 0; i < KDIM * NCOLS / 256; ++i)
        wlds[i * 256 + tid] = W[i * 256 + tid];
    __syncthreads();

    int wave  = tid >> 5;
    int lane  = tid & 31;
    int strip = blockIdx.x * 8 + wave;
    int row_base = strip * 16;
    if (row_base >= nrows) return;        // after the barrier: safe

    int r = lane & 15;
    int g = lane >> 4;                    // 0 or 1
    int row = row_base + r;
    if (row >= nrows) row = nrows - 1;    // clamp tail loads (stores are guarded)

    v8f acc[NCT] = {};

    const float* xrow = X + (size_t)row * KDIM + 2 * g;
    const float* wl   = wlds + 2 * g * NCOLS + r;

    #pragma unroll
    for (int kk = 0; kk < KDIM; kk += 4) {
        v2f a = *(const v2f*)(xrow + kk);            // K = kk+2g, kk+2g+1
        if (RELU) { a.x = fmaxf(a.x, 0.0f); a.y = fmaxf(a.y, 0.0f); }
        #pragma unroll
        for (int t = 0; t < NCT; ++t) {
            v2f b;
            b.x = wl[kk * NCOLS + t * 16];           // W[kk+2g  ][t*16+r]
            b.y = wl[(kk + 1) * NCOLS + t * 16];     // W[kk+2g+1][t*16+r]
            acc[t] = __builtin_amdgcn_wmma_f32_16x16x4_f32(
                false, a, false, b, (short)0, acc[t], false, false);
        }
    }

    #pragma unroll
    for (int t = 0; t < NCT; ++t) {
        int col = t * 16 + r;
        float bv = bias ? bias[col] : 0.0f;
        #pragma unroll
        for (int j = 0; j < 8; ++j) {
            int orow = row_base + j + 8 * g;
            if (orow < nrows) Y[(size_t)orow * NCOLS + col] = acc[t][j] + bv;
        }
    }
}

// ---------------- epilogue: out = h * dinv[node] + b[f] (overwrite) ----------------

__global__ void gcn_self_bias(const float* __restrict__ h, const float* __restrict__ dinv,
                              const float* __restrict__ b, float* __restrict__ out, int n) {
    int i = blockIdx.x * blockDim.x + threadIdx.x;   // over n*64
    if (i < n * HDIM) {
        int node = i >> 6;
        int f    = i & 63;
        out[i] = h[i] * dinv[node] + b[f];
    }
}

// ---------------- edge scatter: out[dst] += h[src] * dis[src]*dis[dst] ----------------
// One wave per edge: 32 lanes x float2 = 64 features. Gather is a coalesced
// 256B read of h[src]'s row; the adds are hardware f32 atomics resolved in L2.

__global__ void gcn_scatter(const int* __restrict__ src, const int* __restrict__ dst,
                            const float* __restrict__ dis, const float* __restrict__ h,
                            float* __restrict__ out, int e) {
    int t  = blockIdx.x * blockDim.x + threadIdx.x;
    int ed = t >> 5;
    if (ed >= e) return;
    int fp = (t & 31) << 1;
    int s = src[ed];
    int d = dst[ed];
    float nrm = dis[s] * dis[d];
    const float2 hv = *(const float2*)(h + (size_t)s * HDIM + fp);
    float* op = out + (size_t)d * HDIM + fp;
    unsafeAtomicAdd(op,     hv.x * nrm);
    unsafeAtomicAdd(op + 1, hv.y * nrm);
}

// ---------------- launch ----------------

extern "C" void kernel_launch(void* const* d_in, const int* in_sizes, int n_in,
                              void* d_out, int out_size, void* d_ws, size_t ws_size,
                              hipStream_t stream) {
    const float* x  = (const float*)d_in[0];
    const int*   ei = (const int*)d_in[1];   // edge_index [2, E]
    const float* Ws = (const float*)d_in[2]; // [3,64,64]
    const float* bs = (const float*)d_in[3]; // [3,64]
    const float* Wc = (const float*)d_in[4]; // [64,32]
    const float* bc = (const float*)d_in[5]; // [32]
    float* out = (float*)d_out;

    const int N = NN, E = NE;
    const int* src = ei;
    const int* dst = ei + E;

    float* dis  = (float*)d_ws;                 // N   (deg, then rsqrt in place)
    float* dinv = dis  + N;                     // N
    float* hbuf = dinv + N;                     // N*64
    float* P    = hbuf + (size_t)N * HDIM;      // N*64
    float* Q    = P    + (size_t)N * HDIM;      // N*64

    // degree + symmetric normalization
    gcn_init_deg <<<(N + 255) / 256, 256, 0, stream>>>(dis, N);
    gcn_deg_accum<<<(E + 255) / 256, 256, 0, stream>>>(dst, dis, E);
    gcn_deg_final<<<(N + 255) / 256, 256, 0, stream>>>(dis, dinv, N);

    const int strips      = (N + 15) / 16;
    const int gemm_blocks = (strips + 7) / 8;       // 8 waves per 256-thread block
    const int ew_blocks   = (E * 32 + 255) / 256;   // wave-per-edge scatter
    const int nv_blocks   = (N * HDIM + 255) / 256;

    const float* cur = x;
    float* pingpong[2] = { P, Q };
    for (int l = 0; l < 3; ++l) {
        // h = relu?(cur) @ Ws[l]   (relu fused into A-load for layers 1,2)
        if (l == 0)
            gcn_gemm_wmma<HDIM, false><<<gemm_blocks, 256, 0, stream>>>(
                cur, Ws + (size_t)l * KDIM * HDIM, nullptr, hbuf, N);
        else
            gcn_gemm_wmma<HDIM, true><<<gemm_blocks, 256, 0, stream>>>(
                cur, Ws + (size_t)l * KDIM * HDIM, nullptr, hbuf, N);
        float* nxt = pingpong[l & 1];
        // nxt = h*dinv + b   (full overwrite -> no zero pass needed)
        gcn_self_bias<<<nv_blocks, 256, 0, stream>>>(hbuf, dinv, bs + (size_t)l * HDIM, nxt, N);
        // nxt[dst] += h[src] * dis[src]*dis[dst]
        gcn_scatter<<<ew_blocks, 256, 0, stream>>>(src, dst, dis, hbuf, nxt, E);
        cur = nxt;
    }

    // classifier: out = relu(cur) @ Wc + bc   (relu fused into A-load)
    gcn_gemm_wmma<ODIM, true><<<gemm_blocks, 256, 0, stream>>>(cur, Wc, bc, out, N);
}